// SurnameClassifier_7241314861503
// MI455X (gfx1250) — compile-verified
//
#include <hip/hip_runtime.h>
#include <hip/hip_bf16.h>
#include <math.h>

typedef float v2f __attribute__((ext_vector_type(2)));
typedef float v8f __attribute__((ext_vector_type(8)));

#define B_ 256
#define S_ 128
#define E_ 512
#define H_ 1024
#define V_ 128
#define C_ 18

// ---------------------------------------------------------------- zero init
__global__ void k_zero(float* __restrict__ p, int n) {
    int i = blockIdx.x * blockDim.x + threadIdx.x;
    if (i < n) p[i] = 0.0f;
}

// ---------------------------------------------------------------------------
// Kernel 1: pre[(s*B+b)*H + h] = dot(emb[x_in[b][s]], W_ih[h]) + b_ih[h]
// GEMM  M = S*B (gathered A rows), K = E, N = H.
// Per wave: one 16x64 output tile (A-fragment reused across 4 column tiles).
// ---------------------------------------------------------------------------
__global__ __launch_bounds__(256)
void k_embed_proj(const int* __restrict__ x_in,
                  const float* __restrict__ emb,
                  const float* __restrict__ W_ih,
                  const float* __restrict__ b_ih,
                  float* __restrict__ pre) {
    const int lane = threadIdx.x & 31;
    const int wave = threadIdx.x >> 5;
    const int gw   = blockIdx.x * (blockDim.x >> 5) + wave;

    const int NG = H_ / 64;                 // 16 column groups of 64
    const int tn = (gw % NG) * 64;
    const int tm = (gw / NG) * 16;

    const int half = lane >> 4;             // selects K pair {0,1} vs {2,3}
    const int l15  = lane & 15;

    // A: gathered embedding row for output row tm+l15
    const int m_a = tm + l15;               // m = s*B + b
    const int s   = m_a / B_;
    const int b   = m_a % B_;
    const int tok = x_in[b * S_ + s];
    const float* __restrict__ arow = emb + (size_t)tok * E_ + half * 2;

    // B: 4 column tiles; B[k][n] = W_ih[n][k]
    const float* __restrict__ brow0 = W_ih + (size_t)(tn +  0 + l15) * E_ + half * 2;
    const float* __restrict__ brow1 = W_ih + (size_t)(tn + 16 + l15) * E_ + half * 2;
    const float* __restrict__ brow2 = W_ih + (size_t)(tn + 32 + l15) * E_ + half * 2;
    const float* __restrict__ brow3 = W_ih + (size_t)(tn + 48 + l15) * E_ + half * 2;

    v8f acc0 = {}; v8f acc1 = {}; v8f acc2 = {}; v8f acc3 = {};

    #pragma unroll 4
    for (int k0 = 0; k0 < E_; k0 += 4) {
        v2f a  = { arow[k0],  arow[k0 + 1] };
        v2f b0 = { brow0[k0], brow0[k0 + 1] };
        v2f b1 = { brow1[k0], brow1[k0 + 1] };
        v2f b2v = { brow2[k0], brow2[k0 + 1] };
        v2f b3 = { brow3[k0], brow3[k0 + 1] };
        acc0 = __builtin_amdgcn_wmma_f32_16x16x4_f32(false, a, false, b0,  (short)0, acc0, false, false);
        acc1 = __builtin_amdgcn_wmma_f32_16x16x4_f32(false, a, false, b1,  (short)0, acc1, false, false);
        acc2 = __builtin_amdgcn_wmma_f32_16x16x4_f32(false, a, false, b2v, (short)0, acc2, false, false);
        acc3 = __builtin_amdgcn_wmma_f32_16x16x4_f32(false, a, false, b3,  (short)0, acc3, false, false);
    }

    // D layout: VGPR r, lane -> (M = r + 8*half, N = l15)
    #pragma unroll
    for (int r = 0; r < 8; ++r) {
        const int row = tm + half * 8 + r;
        float* __restrict__ dst = pre + (size_t)row * H_ + tn + l15;
        dst[ 0] = acc0[r] + b_ih[tn +  0 + l15];
        dst[16] = acc1[r] + b_ih[tn + 16 + l15];
        dst[32] = acc2[r] + b_ih[tn + 32 + l15];
        dst[48] = acc3[r] + b_ih[tn + 48 + l15];
    }
}

// ---------------------------------------------------------------------------
// Kernel 2: one RNN step.
// h_out = tanh(pre_s + h_in @ W_hh^T + b_hh);  if (len[b]-1 == s) last[b] = h_out
// GEMM M = B (256), N = H (1024), K = H (1024).  16x16 tile per wave -> 1024
// waves per step for latency hiding on the serial critical path.
// ---------------------------------------------------------------------------
__global__ __launch_bounds__(256)
void k_rnn_step(const float* __restrict__ hin,
                const float* __restrict__ W_hh,
                const float* __restrict__ b_hh,
                const float* __restrict__ pre_s,
                const int* __restrict__ lengths,
                int s,
                float* __restrict__ hout,
                float* __restrict__ last) {
    const int lane = threadIdx.x & 31;
    const int wave = threadIdx.x >> 5;
    const int gw   = blockIdx.x * (blockDim.x >> 5) + wave;

    const int NT = H_ / 16;                 // 64 column tiles
    const int tn = (gw % NT) * 16;
    const int tm = (gw / NT) * 16;

    const int half = lane >> 4;
    const int l15  = lane & 15;

    const float* __restrict__ arow = hin  + (size_t)(tm + l15) * H_ + half * 2;
    const float* __restrict__ brow = W_hh + (size_t)(tn + l15) * H_ + half * 2;

    v8f acc = {};
    #pragma unroll 8
    for (int k0 = 0; k0 < H_; k0 += 4) {
        v2f a = { arow[k0], arow[k0 + 1] };
        v2f b = { brow[k0], brow[k0 + 1] };
        acc = __builtin_amdgcn_wmma_f32_16x16x4_f32(false, a, false, b, (short)0, acc, false, false);
    }

    const int   col  = tn + l15;
    const float bias = b_hh[col];
    #pragma unroll
    for (int r = 0; r < 8; ++r) {
        const int row = tm + half * 8 + r;        // batch index
        const float v = tanhf(acc[r] + pre_s[(size_t)row * H_ + col] + bias);
        hout[(size_t)row * H_ + col] = v;
        if (lengths[row] - 1 == s) last[(size_t)row * H_ + col] = v;
    }
}

// ---------------------------------------------------------------------------
// Kernel 3: y1 = relu(last @ W1^T + b1).  Same GEMM shape as the RNN step.
// ---------------------------------------------------------------------------
__global__ __launch_bounds__(256)
void k_mlp1(const float* __restrict__ last,
            const float* __restrict__ W1,
            const float* __restrict__ b1,
            float* __restrict__ y1) {
    const int lane = threadIdx.x & 31;
    const int wave = threadIdx.x >> 5;
    const int gw   = blockIdx.x * (blockDim.x >> 5) + wave;

    const int NT = H_ / 16;
    const int tn = (gw % NT) * 16;
    const int tm = (gw / NT) * 16;

    const int half = lane >> 4;
    const int l15  = lane & 15;

    const float* __restrict__ arow = last + (size_t)(tm + l15) * H_ + half * 2;
    const float* __restrict__ brow = W1   + (size_t)(tn + l15) * H_ + half * 2;

    v8f acc = {};
    #pragma unroll 8
    for (int k0 = 0; k0 < H_; k0 += 4) {
        v2f a = { arow[k0], arow[k0 + 1] };
        v2f b = { brow[k0], brow[k0 + 1] };
        acc = __builtin_amdgcn_wmma_f32_16x16x4_f32(false, a, false, b, (short)0, acc, false, false);
    }

    const int   col  = tn + l15;
    const float bias = b1[col];
    #pragma unroll
    for (int r = 0; r < 8; ++r) {
        const int row = tm + half * 8 + r;
        const float v = acc[r] + bias;
        y1[(size_t)row * H_ + col] = v > 0.0f ? v : 0.0f;
    }
}

// ---------------------------------------------------------------------------
// Kernel 4: out[b][c] = dot(y1[b], W2[c]) + b2[c].  Tiny (256x18x1024).
// ---------------------------------------------------------------------------
__global__ void k_mlp2(const float* __restrict__ y1,
                       const float* __restrict__ W2,
                       const float* __restrict__ b2,
                       float* __restrict__ out) {
    int i = blockIdx.x * blockDim.x + threadIdx.x;
    if (i >= B_ * C_) return;
    const int b = i / C_;
    const int c = i % C_;
    const float* __restrict__ yr = y1 + (size_t)b * H_;
    const float* __restrict__ wr = W2 + (size_t)c * H_;
    float acc = b2[c];
    #pragma unroll 8
    for (int k = 0; k < H_; ++k) acc += yr[k] * wr[k];
    out[i] = acc;
}

// ---------------------------------------------------------------------------
extern "C" void kernel_launch(void* const* d_in, const int* in_sizes, int n_in,
                              void* d_out, int out_size, void* d_ws, size_t ws_size,
                              hipStream_t stream) {
    const int*   x_in  = (const int*)  d_in[0];
    const int*   x_len = (const int*)  d_in[1];
    const float* emb   = (const float*)d_in[2];
    const float* W_ih  = (const float*)d_in[3];
    const float* b_ih  = (const float*)d_in[4];
    const float* W_hh  = (const float*)d_in[5];
    const float* b_hh  = (const float*)d_in[6];
    const float* W1    = (const float*)d_in[7];
    const float* b1    = (const float*)d_in[8];
    const float* W2    = (const float*)d_in[9];
    const float* b2    = (const float*)d_in[10];
    float* out = (float*)d_out;

    // workspace layout (floats)
    float* pre  = (float*)d_ws;                         // S*B*H   = 128 MiB
    float* hb0  = pre  + (size_t)S_ * B_ * H_;          // B*H
    float* hb1  = hb0  + (size_t)B_ * H_;               // B*H
    float* last = hb1  + (size_t)B_ * H_;               // B*H
    float* y1   = last + (size_t)B_ * H_;               // B*H

    // h0 = 0
    k_zero<<<(B_ * H_ + 255) / 256, 256, 0, stream>>>(hb0, B_ * H_);

    // Phase 1: embedding + input projection (fully parallel)
    {
        const int waves  = (S_ * B_ / 16) * (H_ / 64);  // 2048 * 16 = 32768
        const int blocks = waves / 8;                   // 256 thr = 8 waves
        k_embed_proj<<<blocks, 256, 0, stream>>>(x_in, emb, W_ih, b_ih, pre);
    }

    // Phase 2: sequential RNN scan, double-buffered h
    {
        const int waves  = (B_ / 16) * (H_ / 16);       // 16 * 64 = 1024
        const int blocks = waves / 8;                   // 128 blocks
        float* bufs[2] = { hb0, hb1 };
        for (int s = 0; s < S_; ++s) {
            const float* hin  = bufs[s & 1];
            float*       hout = bufs[(s + 1) & 1];
            k_rnn_step<<<blocks, 256, 0, stream>>>(
                hin, W_hh, b_hh, pre + (size_t)s * B_ * H_, x_len, s, hout, last);
        }
    }

    // Phase 3: MLP head
    {
        const int waves  = (B_ / 16) * (H_ / 16);
        const int blocks = waves / 8;
        k_mlp1<<<blocks, 256, 0, stream>>>(last, W1, b1, y1);
        k_mlp2<<<(B_ * C_ + 127) / 128, 128, 0, stream>>>(y1, W2, b2, out);
    }
}